// ComplexOscillator_71201967833407
// MI455X (gfx1250) — compile-verified
//
#include <hip/hip_runtime.h>

// CDNA5 (gfx1250) single-pass oscillator-bank phase scan + cos.
// 1024 rows x 96000 samples. One workgroup (8 wave32) per row; each wave
// scans a 256-element sub-tile as a 16x16 f32 matrix using chained
// V_WMMA_F32_16X16X4_F32 triangular matmuls (tensor-core scan).
//
// Phase is accumulated in TURNS (cycles), not radians: the hardware
// transcendental V_COS_F32 computes cos(2*pi*x), so no software range
// reduction is needed -- one TRANS op per output element. Carries are
// wrapped mod 1.0 (exactly periodic), which also keeps precision far
// better than an unwrapped radian phase reaching ~1.5e5.
//
// Traffic = read input once + write output once (~786 MB -> ~34 us @ 23.3 TB/s).

typedef __attribute__((ext_vector_type(2))) float v2f;   // A/B operand: 16x4 / 4x16 f32
typedef __attribute__((ext_vector_type(8))) float v8f;   // C/D operand: 16x16 f32

namespace {
constexpr int   kT        = 96000;   // samples per row
constexpr int   kTile     = 2048;    // per-block tile (8 waves * 256)
constexpr int   kWaveTile = 256;     // per-wave 16x16 matrix
constexpr int   kWaves    = 8;
constexpr int   kIters    = (kT + kTile - 1) / kTile;     // 47 (last tile partial)
constexpr float kNyquist  = 24000.0f;                      // SAMPLE_RATE / 2
constexpr float kTurns    = 1.0f / 48000.0f;               // turns per sample per Hz
constexpr float kInvTwoPi = 0.15915494309189533577f;       // 1 / (2*pi)
}

// Broadcast a value from a compile-time-constant lane: single v_readlane_b32,
// stays off the LDS pipe (vs __shfl -> ds_bpermute) and has no DScnt wait.
__device__ __forceinline__ float bcast_lane(float v, int srclane) {
  return __int_as_float(__builtin_amdgcn_readlane(__float_as_int(v), srclane));
}

__global__ __launch_bounds__(256, 2) void osc_wmma_scan(
    const float* __restrict__ freq,     // [rows][kT]
    const float* __restrict__ phase0,   // [rows] (radians)
    float* __restrict__ out)            // [rows][kT]
{
  __shared__ __align__(16) float lds_tile[kTile];  // staged omega tile (turns)
  __shared__ float lds_tot[kWaves];                // per-wave sub-tile totals

  const int row  = blockIdx.x;
  const int tid  = threadIdx.x;
  const int lane = tid & 31;
  const int wave = tid >> 5;
  const int col  = lane & 15;   // N for B/C/D layouts, M-row for A layout
  const int hi   = lane >> 4;   // lane group (ISA 7.12.2 layouts)

  const float* __restrict__ frow = freq + (size_t)row * kT;
  float*       __restrict__ orow = out  + (size_t)row * kT;

  // ---- constant triangular operands (built per-lane, ISA A/B layouts) ----
  // ub[k] : rows 4k..4k+3 of U (U[r][c] = 1 if r <= c)  -> B operand (4x16)
  // la[k] : cols 4k..4k+3 of Lstrict (L[i][j] = 1 if j < i) -> A operand (16x4)
  v2f ub[4], la[4];
#pragma unroll
  for (int k = 0; k < 4; ++k) {
    const int r0 = 4 * k + 2 * hi;
    ub[k].x = (r0     <= col) ? 1.0f : 0.0f;
    ub[k].y = (r0 + 1 <= col) ? 1.0f : 0.0f;
    la[k].x = (r0     <  col) ? 1.0f : 0.0f;
    la[k].y = (r0 + 1 <  col) ? 1.0f : 0.0f;
  }

  // Running phase carry in TURNS; initial phase (radians) folded in.
  float carry = phase0[row] * kInvTwoPi;

  for (int it = 0; it < kIters; ++it) {
    const int tbase = it * kTile;

    // ---- stage: coalesced float4 loads, mask >= Nyquist, scale to turns ----
    const int gpos = tbase + tid * 8;               // 8 contiguous elems/thread
    float4 x0 = make_float4(0.f, 0.f, 0.f, 0.f), x1 = x0;
    if (gpos + 8 <= kT) {                           // chunks are all-in/all-out
      const float4* gp = (const float4*)(frow + gpos);
      x0 = gp[0];
      x1 = gp[1];
    }
    // prefetch next tile into cache hierarchy (global_prefetch_b8)
    if (gpos + kTile + 8 <= kT)
      __builtin_prefetch(frow + gpos + kTile, 0, 1);

    float4 o0, o1;
    o0.x = (x0.x < kNyquist) ? x0.x * kTurns : 0.0f;
    o0.y = (x0.y < kNyquist) ? x0.y * kTurns : 0.0f;
    o0.z = (x0.z < kNyquist) ? x0.z * kTurns : 0.0f;
    o0.w = (x0.w < kNyquist) ? x0.w * kTurns : 0.0f;
    o1.x = (x1.x < kNyquist) ? x1.x * kTurns : 0.0f;
    o1.y = (x1.y < kNyquist) ? x1.y * kTurns : 0.0f;
    o1.z = (x1.z < kNyquist) ? x1.z * kTurns : 0.0f;
    o1.w = (x1.w < kNyquist) ? x1.w * kTurns : 0.0f;
    ((float4*)lds_tile)[tid * 2 + 0] = o0;
    ((float4*)lds_tile)[tid * 2 + 1] = o1;
    __syncthreads();

    // ---- wave tile scan: 16x16 matrix X, time index = M*16 + N ----
    const int wbase = wave * kWaveTile;

    // Step 1: R = X * U   (row-wise inclusive prefix), K=16 as 4 chained K=4.
    // A operand (16x4 layout): lane holds X[col][4k+2*hi .. +1] -> ds_load_b64.
    v8f r = {0.f, 0.f, 0.f, 0.f, 0.f, 0.f, 0.f, 0.f};
#pragma unroll
    for (int k = 0; k < 4; ++k) {
      const v2f a = *(const v2f*)&lds_tile[wbase + col * 16 + 4 * k + 2 * hi];
      r = __builtin_amdgcn_wmma_f32_16x16x4_f32(
          /*neg_a=*/false, a, /*neg_b=*/false, ub[k],
          /*c_mod=*/(short)0, r, /*reuse_a=*/false, /*reuse_b=*/false);
    }

    // Row totals t[m] = R[m][15]: D layout -> lane 15 (m 0..7, VGPR m) and
    // lane 31 (m 8..15, VGPR m-8). Constant source lanes -> v_readlane_b32.
    float t15[8], t31[8];
#pragma unroll
    for (int d = 0; d < 8; ++d) {
      t15[d] = bcast_lane(r[d], 15);
      t31[d] = bcast_lane(r[d], 31);
    }

    // Step 2: Z = Lstrict * (t . 1^T) + R  (accumulate into R's registers).
    // B operand rows carry t[rowK] broadcast across all 16 columns.
    v8f z = r;
#pragma unroll
    for (int k = 0; k < 4; ++k) {
      const float* src = (k < 2) ? t15 : t31;       // rows 0..7 vs 8..15
      const int    i0  = 4 * (k & 1);               // within-half row offset
      v2f bt;
      bt.x = hi ? src[i0 + 2] : src[i0 + 0];
      bt.y = hi ? src[i0 + 3] : src[i0 + 1];
      z = __builtin_amdgcn_wmma_f32_16x16x4_f32(
          false, la[k], false, bt, (short)0, z, false, false);
    }

    // ---- cross-wave + cross-iteration carry ----
    const float wtot = bcast_lane(z[7], 31);        // Z[15][15] = sub-tile total
    if (lane == 0) lds_tot[wave] = wtot;
    __syncthreads();

    float basep = carry;                            // exclusive prefix for wave
    float all   = 0.0f;
#pragma unroll
    for (int w = 0; w < kWaves; ++w) {
      const float tw = lds_tot[w];
      if (w < wave) basep += tw;
      all += tw;
    }
    // Wrap carry mod 1 turn (cos is 1-turn periodic -> exact).
    const float cnew = carry + all;
    carry = cnew - truncf(cnew);

    // Wrap this wave's base too: per-element argument then stays < ~130 turns,
    // well inside V_COS_F32's valid input domain.
    basep -= truncf(basep);

    // ---- cos + store: D layout rows are 64B-contiguous per VGPR ----
    // V_COS_F32 computes cos(2*pi*x): phase is already in turns, so this is
    // a single TRANS op per element (co-executes with VALU), no libm reduction.
    const int tstart = tbase + wave * kWaveTile;    // sub-tiles all-in/all-out
    if (tstart < kT) {
#pragma unroll
      for (int d = 0; d < 8; ++d) {
        const int m = d + 8 * hi;                   // matrix row M
        orow[tstart + m * 16 + col] = __builtin_amdgcn_cosf(z[d] + basep);
      }
    }
  }
}

extern "C" void kernel_launch(void* const* d_in, const int* in_sizes, int n_in,
                              void* d_out, int out_size, void* d_ws, size_t ws_size,
                              hipStream_t stream) {
  const float* freq   = (const float*)d_in[0];   // [B, N, T] f32
  const float* phase0 = (const float*)d_in[1];   // [B, N, 1] f32
  float*       out    = (float*)d_out;           // [B, N, T] f32
  const int rows = in_sizes[1];                  // B*N = 1024 (one block each)
  osc_wmma_scan<<<dim3(rows), dim3(256), 0, stream>>>(freq, phase0, out);
}